// SpatialTransformer_5600637354401
// MI455X (gfx1250) — compile-verified
//
#include <hip/hip_runtime.h>
#include <hip/hip_bf16.h>

// 3D spatial transformer (trilinear warp, clamp-to-edge), float32.
// vol:  [B, D, H, W, 1]  flow: [B, D, H, W, 3]  out: [B, D, H, W, 1]
//
// MI455X strategy: memory-bound gather (~200 MB min traffic -> ~9 us floor
// at 23.3 TB/s). vol (39 MB) fits in the 192 MB global L2, so vol gathers
// stay temporal (L2-resident) while the read-once flow stream and the
// write-once output use non-temporal cache policy (th:TH_NT on gfx1250) to
// avoid evicting vol. 3D exact-tiling launch removes all per-thread integer
// division; each wave32 covers 32 contiguous W-voxels (coalesced b96 flow
// load, coalesced NT b32 store, compact gather footprint).

namespace {

constexpr int kB = 2;
constexpr int kD = 160;
constexpr int kH = 192;
constexpr int kW = 160;
constexpr int kHW  = kH * kW;
constexpr int kDHW = kD * kHW;

constexpr int kBx = 32;   // one wave per W-run
constexpr int kBy = 8;    // 8 waves per block

struct Axis {
  float w_lo;  // weight of lower corner (reference "diff1")
  float w_hi;  // weight of upper corner (reference "diff0")
  int   i0;    // lower index (clipped)
  int   i1;    // upper index (clipped)
};

__device__ __forceinline__ Axis axis_interp(float coord, float disp, float maxv) {
  // Reference semantics (neurite/voxelmorph):
  //   loc       = grid + flow
  //   loc_clip  = clip(loc, 0, max)
  //   loc0      = floor(loc)                (UNclipped floor)
  //   loc0_clip = clip(loc0, 0, max)
  //   loc1_clip = clip(loc0_clip + 1, 0, max)
  //   diff1     = loc1_clip - loc_clip      (lower-corner weight)
  //   diff0     = 1 - diff1                 (upper-corner weight)
  float loc      = coord + disp;
  float loc_clip = fminf(fmaxf(loc, 0.0f), maxv);
  float loc0     = floorf(loc);
  float loc0c    = fminf(fmaxf(loc0, 0.0f), maxv);
  float loc1c    = fminf(loc0c + 1.0f, maxv);   // loc0c >= 0, no lower clip needed
  Axis a;
  a.w_lo = loc1c - loc_clip;
  a.w_hi = 1.0f - a.w_lo;
  a.i0   = (int)loc0c;
  a.i1   = (int)loc1c;
  return a;
}

__global__ __launch_bounds__(kBx * kBy) void warp3d_trilinear_kernel(
    const float* __restrict__ vol,
    const float* __restrict__ flow,
    float* __restrict__ out) {
  // Exact tiling: no bounds checks, no per-thread div/mod.
  const int x  = blockIdx.x * kBx + threadIdx.x;   // 0..159
  const int y  = blockIdx.y * kBy + threadIdx.y;   // 0..191
  const int zb = blockIdx.z;                       // b*kD + z, 0..319
  const int z  = zb % kD;                          // wave-uniform -> SALU
  const int b  = zb / kD;                          // wave-uniform -> SALU

  const int idx = (zb * kH + y) * kW + x;          // global flat voxel index

  // Streaming read of the displacement vector: non-temporal (read-once).
  const float* fp = flow + (size_t)idx * 3;
  const float fz = __builtin_nontemporal_load(fp + 0);  // D-axis displacement
  const float fy = __builtin_nontemporal_load(fp + 1);  // H-axis displacement
  const float fx = __builtin_nontemporal_load(fp + 2);  // W-axis displacement

  const Axis az = axis_interp((float)z, fz, (float)(kD - 1));
  const Axis ay = axis_interp((float)y, fy, (float)(kH - 1));
  const Axis ax = axis_interp((float)x, fx, (float)(kW - 1));

  const float* vb = vol + (size_t)b * kDHW;
  const int z0 = az.i0 * kHW, z1 = az.i1 * kHW;
  const int y0 = ay.i0 * kW,  y1 = ay.i1 * kW;

  // 8 gather loads; vol is L2-resident (39 MB << 192 MB L2), keep temporal.
  const float v000 = vb[z0 + y0 + ax.i0];
  const float v001 = vb[z0 + y0 + ax.i1];
  const float v010 = vb[z0 + y1 + ax.i0];
  const float v011 = vb[z0 + y1 + ax.i1];
  const float v100 = vb[z1 + y0 + ax.i0];
  const float v101 = vb[z1 + y0 + ax.i1];
  const float v110 = vb[z1 + y1 + ax.i0];
  const float v111 = vb[z1 + y1 + ax.i1];

  // Weight product order matches reference: ((w_z * w_y) * w_x).
  const float wz0y0 = az.w_lo * ay.w_lo;
  const float wz0y1 = az.w_lo * ay.w_hi;
  const float wz1y0 = az.w_hi * ay.w_lo;
  const float wz1y1 = az.w_hi * ay.w_hi;

  float acc;
  acc  = (wz0y0 * ax.w_lo) * v000;
  acc += (wz0y0 * ax.w_hi) * v001;
  acc += (wz0y1 * ax.w_lo) * v010;
  acc += (wz0y1 * ax.w_hi) * v011;
  acc += (wz1y0 * ax.w_lo) * v100;
  acc += (wz1y0 * ax.w_hi) * v101;
  acc += (wz1y1 * ax.w_lo) * v110;
  acc += (wz1y1 * ax.w_hi) * v111;

  // Streaming write-once result: non-temporal store.
  __builtin_nontemporal_store(acc, out + idx);
}

}  // namespace

extern "C" void kernel_launch(void* const* d_in, const int* in_sizes, int n_in,
                              void* d_out, int out_size, void* d_ws, size_t ws_size,
                              hipStream_t stream) {
  (void)in_sizes; (void)n_in; (void)d_ws; (void)ws_size; (void)out_size;
  const float* vol  = (const float*)d_in[0];
  const float* flow = (const float*)d_in[1];
  float* out = (float*)d_out;

  static_assert(kW % kBx == 0 && kH % kBy == 0, "exact tiling required");
  const dim3 block(kBx, kBy, 1);                 // 256 threads = 8 wave32
  const dim3 grid(kW / kBx, kH / kBy, kB * kD);  // (5, 24, 320)
  warp3d_trilinear_kernel<<<grid, block, 0, stream>>>(vol, flow, out);
}